// BayesianSTDPAdaptive_30116310680365
// MI455X (gfx1250) — compile-verified
//
#include <hip/hip_runtime.h>
#include <hip/hip_bf16.h>

// Fused Bayesian-STDP adaptive update for MI455X (gfx1250, wave32).
// Each wave owns a 16x16 (O x I) weight tile in the native WMMA f32 C/D
// layout; per scan step it computes its corr tile with 3x
// v_wmma_f32_16x16x4_f32 (K=10 padded to 12) and applies the elementwise
// adaptive-learning-rate update entirely in registers.

typedef float v2f __attribute__((ext_vector_type(2)));
typedef float v8f __attribute__((ext_vector_type(8)));

#define T_ITERS   200
#define TB        10
#define I_DIM     1000
#define O_DIM     100
#define BASE_MU_W 1.0f
#define BASE_MU_B 0.5f
#define MIN_MU_W  1e-6f
#define MIN_MU_B  1e-6f
#define MAX_DELTA 10.0f
#define MOM       0.001f

__device__ __forceinline__ float clip_pm(float x, float lim) {
    // jnp.clip(x, -lim, lim); +/-Inf clamp correctly via fmin/fmax.
    return fminf(fmaxf(x, -lim), lim);
}

__global__ __launch_bounds__(32) void stdp_weights_kernel(
    const float* __restrict__ psp,   // (2000, 1000) input_psp
    const float* __restrict__ spk,   // (2000, 100)  output_spikes
    const float* __restrict__ w_in,  // (100, 1000)  weights
    float* __restrict__ out)         // out[0 .. 100000) = final weights
{
    const int  lane  = threadIdx.x & 31;
    const int  col   = lane & 15;
    const bool lo    = lane < 16;            // lower half-wave
    const int  i_tile = blockIdx.x;          // 0..62
    const int  o_tile = blockIdx.y;          // 0..6
    const int  i_col  = i_tile * 16 + col;   // N column (i) this lane owns
    const int  o_colA = o_tile * 16 + col;   // M column this lane loads for A/tot
    const bool i_ok   = (i_col < I_DIM);
    const bool o_okA  = (o_colA < O_DIM);
    const int  o_row0 = o_tile * 16 + (lo ? 0 : 8);  // row of slot r is o_row0 + r

    // Per-slot adaptive-LR state, resident in VGPRs (WMMA C/D layout).
    float w[8], mu_w[8], w1[8], w2[8];
#pragma unroll
    for (int r = 0; r < 8; ++r) {
        const int o_r = o_row0 + r;
        const float wv = (o_r < O_DIM && i_ok) ? w_in[o_r * I_DIM + i_col] : 0.0f;
        w[r]  = wv;
        mu_w[r] = BASE_MU_W;
        w1[r] = wv;
        w2[r] = wv * wv + BASE_MU_W;
    }

    const int bperm_idx = ((lane & 16) >> 1);  // +8 rows for the upper half-wave

    for (int t = 0; t < T_ITERS; ++t) {
        // Spike column for this lane's A-column o: s[b] = spikes[t, b, o].
        // These feed both the WMMA A operand and total_out[o].
        float s[TB];
        float tot = 0.0f;
#pragma unroll
        for (int b = 0; b < TB; ++b) {
            const float v = o_okA ? spk[(t * TB + b) * O_DIM + o_colA] : 0.0f;
            s[b] = v;
            tot += v;
        }

        // PSP values for the B operand (K split across half-waves: lo={0,1},hi={2,3}).
        const float* pr = psp + (size_t)(t * TB) * I_DIM + i_col;
        const int boff = lo ? 0 : 2;
        const float p0 = i_ok ? pr[(0 + boff) * I_DIM] : 0.0f;
        const float p1 = i_ok ? pr[(1 + boff) * I_DIM] : 0.0f;
        const float p2 = i_ok ? pr[(4 + boff) * I_DIM] : 0.0f;
        const float p3 = i_ok ? pr[(5 + boff) * I_DIM] : 0.0f;
        const float p4 = (i_ok && lo) ? pr[8 * I_DIM] : 0.0f;  // b=10,11 padded 0
        const float p5 = (i_ok && lo) ? pr[9 * I_DIM] : 0.0f;

        // corr tile: D = A(16x4 spikes^T) x B(4x16 psps), K accumulated 0..11.
        v8f acc = {};
        v2f A, B;
        A.x = lo ? s[0] : s[2];  A.y = lo ? s[1] : s[3];
        B.x = p0;                B.y = p1;
        acc = __builtin_amdgcn_wmma_f32_16x16x4_f32(false, A, false, B,
                                                    (short)0, acc, false, false);
        A.x = lo ? s[4] : s[6];  A.y = lo ? s[5] : s[7];
        B.x = p2;                B.y = p3;
        acc = __builtin_amdgcn_wmma_f32_16x16x4_f32(false, A, false, B,
                                                    (short)0, acc, false, false);
        A.x = lo ? s[8] : 0.0f;  A.y = lo ? s[9] : 0.0f;   // K=10,11 are zero pad
        B.x = p4;                B.y = p5;
        acc = __builtin_amdgcn_wmma_f32_16x16x4_f32(false, A, false, B,
                                                    (short)0, acc, false, false);

        // Elementwise adaptive update per slot.
#pragma unroll
        for (int r = 0; r < 8; ++r) {
            // total_out[t, o_row0 + r]: broadcast from lane (r) / (r+8).
            const float tot_r = __int_as_float(__builtin_amdgcn_ds_bpermute(
                (r + bperm_idx) << 2, __float_as_int(tot)));
            float bump = acc[r] * __expf(-w[r]);       // corr * C * exp(-w)
            if (bump != bump) bump = 0.0f;             // nan_to_num
            const float dw = clip_pm(bump - tot_r, MAX_DELTA);
            const float wn = w[r] + mu_w[r] * dw;
            w[r]  = wn;
            w1[r] = MOM * wn + (1.0f - MOM) * w1[r];
            w2[r] = MOM * wn * wn + (1.0f - MOM) * w2[r];
            const float mu = (w2[r] - w1[r] * w1[r]) / (__expf(-w1[r]) + 1.0f);
            mu_w[r] = fminf(fmaxf(mu, MIN_MU_W), BASE_MU_W);
        }
    }

#pragma unroll
    for (int r = 0; r < 8; ++r) {
        const int o_r = o_row0 + r;
        if (o_r < O_DIM && i_ok) out[o_r * I_DIM + i_col] = w[r];
    }
}

__global__ __launch_bounds__(128) void stdp_bias_kernel(
    const float* __restrict__ spk,   // (2000, 100)
    const float* __restrict__ b_in,  // (100,)
    float* __restrict__ out)         // out[100000 .. 100100) = final biases
{
    __shared__ float ts[T_ITERS];    // total_out_sum[t]
    const int tid = threadIdx.x;

    for (int t = tid; t < T_ITERS; t += blockDim.x) {
        const float* p = spk + (size_t)t * TB * O_DIM;
        float ssum = 0.0f;
        for (int k = 0; k < TB * O_DIM; ++k) ssum += p[k];
        ts[t] = ssum;
    }
    __syncthreads();

    if (tid >= O_DIM) return;

    float b  = b_in[tid];
    float mu = BASE_MU_B;
    float b1 = b;
    float b2 = b * b + BASE_MU_B;

    for (int t = 0; t < T_ITERS; ++t) {
        float tot = 0.0f;
#pragma unroll
        for (int bb = 0; bb < TB; ++bb)
            tot += spk[(t * TB + bb) * O_DIM + tid];
        float bump = __expf(-b) * tot;
        if (bump != bump) bump = 0.0f;
        const float db = clip_pm(bump - ts[t], MAX_DELTA);
        b  = b + mu * db;
        b1 = MOM * b + (1.0f - MOM) * b1;
        b2 = MOM * b * b + (1.0f - MOM) * b2;
        const float m = (b2 - b1 * b1) / (__expf(-b1) + 1.0f);
        mu = fminf(fmaxf(m, MIN_MU_B), BASE_MU_B);
    }
    out[O_DIM * I_DIM + tid] = b;
}

extern "C" void kernel_launch(void* const* d_in, const int* in_sizes, int n_in,
                              void* d_out, int out_size, void* d_ws, size_t ws_size,
                              hipStream_t stream) {
    (void)in_sizes; (void)n_in; (void)out_size; (void)d_ws; (void)ws_size;
    const float* psp  = (const float*)d_in[0];   // input_psp   (2000,1000)
    const float* spk  = (const float*)d_in[1];   // output_spikes (2000,100)
    const float* w_in = (const float*)d_in[2];   // weights (100,1000)
    const float* b_in = (const float*)d_in[3];   // biases (100,)
    float* out = (float*)d_out;                  // [w | b] concatenated

    dim3 grid(63, 7);                            // 63 i-tiles x 7 o-tiles, 1 wave each
    stdp_weights_kernel<<<grid, 32, 0, stream>>>(psp, spk, w_in, out);
    stdp_bias_kernel<<<1, 128, 0, stream>>>(spk, b_in, out);
}